// Aggregationfeature_22995254903081
// MI455X (gfx1250) — compile-verified
//
#include <hip/hip_runtime.h>

// Problem constants (from reference)
#define T_ 11
#define N_ 20000
#define E_ 200000
#define R_ 50000
#define IN_ 64
#define H_ 128
#define C_ 2

typedef float v2f __attribute__((ext_vector_type(2)));
typedef float v8f __attribute__((ext_vector_type(8)));

// ---------------------------------------------------------------- utilities

// Vectorized zero fill: n must be a multiple of 4.
__global__ __launch_bounds__(256) void zero_f32v4(float* p, int n4) {
    int i = blockIdx.x * 256 + threadIdx.x;
    if (i < n4) ((float4*)p)[i] = make_float4(0.f, 0.f, 0.f, 0.f);
}

__global__ __launch_bounds__(256) void zero_f32(float* p, int n) {
    int i = blockIdx.x * 256 + threadIdx.x;
    if (i < n) p[i] = 0.0f;
}

// Count out-degree (by edge_src) and in-degree (by edge_dst) per slice.
__global__ __launch_bounds__(256) void deg_count(const int* __restrict__ src,
                                                 const int* __restrict__ dst,
                                                 float* __restrict__ cnt_out,
                                                 float* __restrict__ cnt_in) {
    int i = blockIdx.x * 256 + threadIdx.x;      // over T*E
    if (i >= T_ * E_) return;
    int t = i / E_;
    atomicAdd(&cnt_out[t * N_ + src[i]], 1.0f);
    atomicAdd(&cnt_in [t * N_ + dst[i]], 1.0f);
}

// In place: p = rsqrt(max(p, 1))
__global__ __launch_bounds__(256) void deg_rsqrt(float* p, int n) {
    int i = blockIdx.x * 256 + threadIdx.x;
    if (i < n) p[i] = rsqrtf(fmaxf(p[i], 1.0f));
}

// ------------------------------------------------- WMMA FP32 GEMM (M x K) @ (K x 128)
//
// Block: 256 threads = 8 waves. Block computes a 16-row stripe across all 128
// output columns; wave w owns the 16x16 tile at columns [16w, 16w+16).
// A tile (16 x KDIM, optionally row-scaled) is staged through LDS once per block.
// B fragments are read directly from global (W is tiny; L2/WGP$ resident).
// Uses V_WMMA_F32_16X16X4_F32, accumulating over KDIM in steps of 4.
//
// Fragment layouts per CDNA5 ISA 7.12.2:
//  A 16x4 : lanes 0-15 -> M=lane,  v0=K0, v1=K1 ; lanes 16-31 -> M=lane-16, v0=K2, v1=K3
//  B 4x16 : v0 -> rows K0 (lanes 0-15) / K1 (lanes 16-31); v1 -> K2 / K3
//  C/D    : VGPR v -> row v (lanes 0-15) / row v+8 (lanes 16-31), col = lane&15
template<int KDIM, bool RELU, bool HAS_SCALE, bool HAS_BIAS>
__global__ __launch_bounds__(256) void gemm_wmma(const float* __restrict__ A,
                                                 const float* __restrict__ W,
                                                 const float* __restrict__ bias,
                                                 const float* __restrict__ scale,
                                                 float* __restrict__ Out) {
    __shared__ float Alds[16][KDIM];
    const int row0 = blockIdx.x * 16;

    // Stage (scaled) A tile into LDS: 16*KDIM elements by 256 threads.
    for (int idx = threadIdx.x; idx < 16 * KDIM; idx += 256) {
        int r = idx / KDIM, k = idx % KDIM;
        float v = A[(row0 + r) * KDIM + k];
        if (HAS_SCALE) v *= scale[row0 + r];
        Alds[r][k] = v;
    }
    __syncthreads();

    const int wave  = threadIdx.x >> 5;
    const int lane  = threadIdx.x & 31;
    const int col0  = wave * 16;
    const int lhalf = lane >> 4;   // 0: lanes 0-15, 1: lanes 16-31
    const int l15   = lane & 15;

    v8f c = {};
    for (int k = 0; k < KDIM; k += 4) {
        v2f a, b;
        a.x = Alds[l15][k + 2 * lhalf + 0];
        a.y = Alds[l15][k + 2 * lhalf + 1];
        b.x = W[(k + lhalf)     * H_ + col0 + l15];
        b.y = W[(k + 2 + lhalf) * H_ + col0 + l15];
        c = __builtin_amdgcn_wmma_f32_16x16x4_f32(false, a, false, b,
                                                  (short)0, c, false, false);
    }

    #pragma unroll
    for (int v = 0; v < 8; ++v) {
        int row = row0 + v + 8 * lhalf;
        int col = col0 + l15;
        float val = c[v];
        if (HAS_BIAS) val += bias[col];
        if (RELU)     val = fmaxf(val, 0.0f);
        Out[row * H_ + col] = val;
    }
}

// ------------------------------------------------- graph scatter / finalize

// agg[d_idx[e]][:] += x[s_idx[e]][:]
// Wave per edge; each lane covers 4 features (32*4 = 128). x read as b128;
// index loads amortized 4x vs thread-per-element. agg is L2-resident (10 MB).
__global__ __launch_bounds__(256) void edge_scatter(const float* __restrict__ x,
                                                    const int* __restrict__ s_idx,
                                                    const int* __restrict__ d_idx,
                                                    float* __restrict__ agg) {
    int e    = (blockIdx.x * 256 + threadIdx.x) >> 5;
    int lane = threadIdx.x & 31;
    if (e >= E_) return;
    int s = s_idx[e];
    int d = d_idx[e];
    float4 v = ((const float4*)(x + s * H_))[lane];
    float* dst = agg + d * H_ + lane * 4;
    atomicAdd(dst + 0, v.x);
    atomicAdd(dst + 1, v.y);
    atomicAdd(dst + 2, v.z);
    atomicAdd(dst + 3, v.w);
}

// out[n][j] = act(agg[n][j] * rsq_in[n] + b[j])   (float4 per thread)
template<bool RELU>
__global__ __launch_bounds__(256) void conv_finalize(const float* __restrict__ agg,
                                                     const float* __restrict__ rsq_in,
                                                     const float* __restrict__ b,
                                                     float* __restrict__ out) {
    int i = blockIdx.x * 256 + threadIdx.x;      // over N*H/4
    if (i >= N_ * H_ / 4) return;
    int n  = i >> 5;                 // H/4 = 32 float4 per node
    int j4 = i & 31;
    float  s  = rsq_in[n];
    float4 a  = ((const float4*)agg)[i];
    const float* bj = b + j4 * 4;
    float4 v;
    v.x = a.x * s + bj[0];
    v.y = a.y * s + bj[1];
    v.z = a.z * s + bj[2];
    v.w = a.w * s + bj[3];
    if (RELU) {
        v.x = fmaxf(v.x, 0.f); v.y = fmaxf(v.y, 0.f);
        v.z = fmaxf(v.z, 0.f); v.w = fmaxf(v.w, 0.f);
    }
    ((float4*)out)[i] = v;
}

// ------------------------------------------------- masked temporal gather-mean

// One wave per row; lane covers 4 features via float4 (32*4 = 128).
__global__ __launch_bounds__(256) void gather_mean(const float* __restrict__ h2,
                                                   const int* __restrict__ row_idx,
                                                   const unsigned char* __restrict__ mask,
                                                   float* __restrict__ mean) {
    int row  = (blockIdx.x * 256 + threadIdx.x) >> 5;
    int lane = threadIdx.x & 31;
    if (row >= R_) return;
    float4 acc = make_float4(0.f, 0.f, 0.f, 0.f);
    float cnt = 0.0f;
    for (int t = 0; t < T_; ++t) {
        if (mask[row * T_ + t]) {
            int n = row_idx[row * T_ + t];
            const float4* p = (const float4*)(h2 + (t * N_ + n) * H_);
            float4 v = p[lane];
            acc.x += v.x; acc.y += v.y; acc.z += v.z; acc.w += v.w;
            cnt += 1.0f;
        }
    }
    float inv = 1.0f / cnt;      // mask[:,0] is always true -> cnt >= 1
    float4 r  = make_float4(acc.x * inv, acc.y * inv, acc.z * inv, acc.w * inv);
    ((float4*)(mean + row * H_))[lane] = r;
}

// ------------------------------------------------- MLP head (128 -> 2)

__global__ __launch_bounds__(256) void mlp_head(const float* __restrict__ y,
                                               const float* __restrict__ W3,
                                               const float* __restrict__ b3,
                                               float* __restrict__ out) {
    int i = blockIdx.x * 256 + threadIdx.x;      // over R*C
    if (i >= R_ * C_) return;
    int row = i / C_, c = i % C_;
    const float* yr = y + row * H_;
    float acc = b3[c];
    #pragma unroll 8
    for (int k = 0; k < H_; ++k) acc += yr[k] * W3[k * C_ + c];
    out[i] = acc;
}

// ---------------------------------------------------------------- launcher

extern "C" void kernel_launch(void* const* d_in, const int* in_sizes, int n_in,
                              void* d_out, int out_size, void* d_ws, size_t ws_size,
                              hipStream_t stream) {
    (void)in_sizes; (void)n_in; (void)out_size; (void)ws_size;

    const float*         node_feat = (const float*)d_in[0];
    const int*           edge_src  = (const int*)d_in[1];
    const int*           edge_dst  = (const int*)d_in[2];
    const int*           row_idx   = (const int*)d_in[3];
    const unsigned char* row_mask  = (const unsigned char*)d_in[4];
    const float*         W_in      = (const float*)d_in[5];
    const float*         b_in      = (const float*)d_in[6];
    const float*         W1        = (const float*)d_in[7];
    const float*         b1        = (const float*)d_in[8];
    const float*         W2        = (const float*)d_in[9];
    const float*         b2        = (const float*)d_in[10];
    const float*         mlp_W1    = (const float*)d_in[11];
    const float*         mlp_b1    = (const float*)d_in[12];
    const float*         mlp_W2    = (const float*)d_in[13];
    const float*         mlp_b2    = (const float*)d_in[14];
    const float*         mlp_W3    = (const float*)d_in[15];
    const float*         mlp_b3    = (const float*)d_in[16];
    float* out = (float*)d_out;

    // Workspace layout (floats); h2 aliases h0, MLP temporaries alias h1.
    float* ws      = (float*)d_ws;
    float* h0      = ws;                             // T*N*H
    float* h1      = h0 + (size_t)T_ * N_ * H_;      // T*N*H
    float* xbuf    = h1 + (size_t)T_ * N_ * H_;      // N*H
    float* agg     = xbuf + (size_t)N_ * H_;         // N*H
    float* rsq_out = agg + (size_t)N_ * H_;          // T*N
    float* rsq_in  = rsq_out + (size_t)T_ * N_;      // T*N (contiguous with rsq_out)
    float* mean    = rsq_in + (size_t)T_ * N_;       // R*H
    float* h2      = h0;                             // reuse: h0 dead after conv1
    float* y1      = h1;                             // reuse: h1 dead after conv2
    float* y2      = h1 + (size_t)R_ * H_;           // fits: 2*R*H < T*N*H

    const int B = 256;

    // 1) Degrees -> rsqrt(max(deg,1)); rsq_out/rsq_in are contiguous.
    zero_f32<<<(2 * T_ * N_ + B - 1) / B, B, 0, stream>>>(rsq_out, 2 * T_ * N_);
    deg_count<<<(T_ * E_ + B - 1) / B, B, 0, stream>>>(edge_src, edge_dst, rsq_out, rsq_in);
    deg_rsqrt<<<(2 * T_ * N_ + B - 1) / B, B, 0, stream>>>(rsq_out, 2 * T_ * N_);

    // 2) Input projection: h0 = relu(node_feat @ W_in + b_in), M = T*N.
    gemm_wmma<IN_, true, false, true><<<(T_ * N_) / 16, B, 0, stream>>>(
        node_feat, W_in, b_in, nullptr, h0);

    // 3) Two hetero-conv passes.
    for (int pass = 0; pass < 2; ++pass) {
        const float* hin  = (pass == 0) ? h0 : h1;
        float*       hout = (pass == 0) ? h1 : h2;
        const float* Wc   = (pass == 0) ? W1 : W2;
        const float* bc   = (pass == 0) ? b1 : b2;
        for (int r = 0; r < T_; ++r) {
            int dst_t = (r + 1) % T_;
            // x = (h[r] * out_deg^-0.5) @ W[r]
            gemm_wmma<H_, false, true, false><<<N_ / 16, B, 0, stream>>>(
                hin + (size_t)r * N_ * H_, Wc + (size_t)r * H_ * H_,
                nullptr, rsq_out + (size_t)r * N_, xbuf);
            // agg = segment_sum(x[s_idx], d_idx)
            zero_f32v4<<<(N_ * H_ / 4 + B - 1) / B, B, 0, stream>>>(agg, N_ * H_ / 4);
            edge_scatter<<<(E_ * 32 + B - 1) / B, B, 0, stream>>>(
                xbuf, edge_src + (size_t)r * E_, edge_dst + (size_t)r * E_, agg);
            // h_out[dst_t] = act(agg * in_deg^-0.5 + b[r])
            if (pass == 0)
                conv_finalize<true><<<(N_ * H_ / 4 + B - 1) / B, B, 0, stream>>>(
                    agg, rsq_in + (size_t)r * N_, bc + (size_t)r * H_,
                    hout + (size_t)dst_t * N_ * H_);
            else
                conv_finalize<false><<<(N_ * H_ / 4 + B - 1) / B, B, 0, stream>>>(
                    agg, rsq_in + (size_t)r * N_, bc + (size_t)r * H_,
                    hout + (size_t)dst_t * N_ * H_);
        }
    }

    // 4) Masked temporal gather-mean: mean[R][H].
    gather_mean<<<(R_ * 32 + B - 1) / B, B, 0, stream>>>(h2, row_idx, row_mask, mean);

    // 5) MLP head.
    gemm_wmma<H_, true, false, true><<<R_ / 16, B, 0, stream>>>(
        mean, mlp_W1, mlp_b1, nullptr, y1);
    gemm_wmma<H_, true, false, true><<<R_ / 16, B, 0, stream>>>(
        y1, mlp_W2, mlp_b2, nullptr, y2);
    mlp_head<<<(R_ * C_ + B - 1) / B, B, 0, stream>>>(y2, mlp_W3, mlp_b3, out);
}